// SURE_pure_78426102825223
// MI455X (gfx1250) — compile-verified
//
#include <hip/hip_runtime.h>
#include <hip/hip_bf16.h>
#include <math.h>

#define EPS 1e-06f
#define KTOT 32768      // F*C*H*W
#define SPA  16384      // C*H*W
#define NC   10
#define ND   4
#define NB   64

typedef float v2f __attribute__((ext_vector_type(2)));
typedef float v8f __attribute__((ext_vector_type(8)));

__device__ __forceinline__ float lsig(float x) {
  // numerically stable log(sigmoid(x))
  return fminf(x, 0.0f) - log1pf(expf(-fabsf(x)));
}

// ---------------------------------------------------------------------------
// Kernel 1: segment-sum bins + counts, x_bins = bins/counts, lsA = ls(x_bins),
// lsM = ls(miu). One thread per feature position p (coalesced across b rows).
// Also zeroes the norm accumulator buffer (runs before the WMMA kernel in
// stream order).
// ---------------------------------------------------------------------------
__global__ __launch_bounds__(256) void bins_kernel(
    const float* __restrict__ x_LE, const int* __restrict__ labels,
    const float* __restrict__ X_LEs, const float* __restrict__ X_weights,
    const float* __restrict__ miu,
    float* __restrict__ x_bins, float* __restrict__ lsA,
    float* __restrict__ lsM, float* __restrict__ normbuf) {
  __shared__ int lab[NB];
  if (threadIdx.x < NB) lab[threadIdx.x] = labels[threadIdx.x];
  if (blockIdx.x == 0 && threadIdx.x < 64) normbuf[threadIdx.x] = 0.0f;
  __syncthreads();

  int p = blockIdx.x * blockDim.x + threadIdx.x;
  if (p >= KTOT) return;

  float acc[NC];
  float cnt[NC];
#pragma unroll
  for (int c = 0; c < NC; ++c) { acc[c] = 0.0f; cnt[c] = 0.0f; }

  for (int b = 0; b < NB; ++b) {
    float v = x_LE[(size_t)b * KTOT + p];
    int l = lab[b];
#pragma unroll
    for (int c = 0; c < NC; ++c) {
      acc[c] += (l == c) ? v : 0.0f;
      cnt[c] += (l == c) ? 1.0f : 0.0f;
    }
  }
#pragma unroll
  for (int c = 0; c < NC; ++c) {
    float cf = X_weights[c] + cnt[c];
    float xb = (X_LEs[(size_t)c * KTOT + p] + acc[c]) / cf;
    x_bins[(size_t)c * KTOT + p] = xb;
    lsA[(size_t)c * KTOT + p] = lsig(xb);
  }
#pragma unroll
  for (int d = 0; d < ND; ++d)
    lsM[(size_t)d * KTOT + p] = lsig(miu[(size_t)d * KTOT + p]);
}

// ---------------------------------------------------------------------------
// Kernel 2: norm[d,c] = sum_k (lsA[c,k]-lsM[d,k])^2 via V_WMMA_F32_16X16X4_F32.
// Augmented operands per real k: A-side (a^2, a, 1), B-side (1, -2m, m^2)
// => accumulator gains exactly (a-m)^2 per k. 3 WMMAs per 4 real k.
//
// Padding trick: A-row m only feeds D-row m; B-col n only feeds D-col n.
// We discard D rows >= NC and cols >= ND, so padding lanes just need a VALID
// address (clamped to row 0) — no zeroing, no EXEC divergence in the loop.
// One wave per block; K split over 256 blocks; atomic tile reduction.
// ---------------------------------------------------------------------------
__global__ __launch_bounds__(32) void norm_wmma_kernel(
    const float* __restrict__ lsA, const float* __restrict__ lsM,
    float* __restrict__ normbuf) {
  const int lane = threadIdx.x;      // 0..31, full wave, no divergence at WMMA
  const int m = lane & 15;           // row (c) for A / col (d) for B
  const bool hi = lane >= 16;        // K sub-pair 2,3 vs 0,1
  const int kPerBlk = KTOT / gridDim.x;    // 128 with 256 blocks
  const int k0 = blockIdx.x * kPerBlk;

  // clamped row pointers: garbage rows/cols land only in discarded D elements
  const float* Arow = lsA + (size_t)(m < NC ? m : 0) * KTOT;
  const float* Mrow = lsM + (size_t)(m < ND ? m : 0) * KTOT;

  v8f acc = {0.f, 0.f, 0.f, 0.f, 0.f, 0.f, 0.f, 0.f};

  for (int r = k0; r < k0 + kPerBlk; r += 4) {
    float4 av = *reinterpret_cast<const float4*>(Arow + r);
    float4 mv = *reinterpret_cast<const float4*>(Mrow + r);

    v2f A, B;
    // ---- step 0: virtual j = 12t+{0..3} ----
    A.x = hi ? 1.0f        : av.x * av.x;
    A.y = hi ? av.y * av.y : av.x;
    B.x = hi ? mv.x * mv.x : 1.0f;
    B.y = hi ? 1.0f        : -2.0f * mv.x;
    acc = __builtin_amdgcn_wmma_f32_16x16x4_f32(false, A, false, B,
                                                (short)0, acc, false, false);
    // ---- step 1: virtual j = 12t+{4..7} ----
    A.x = hi ? av.z * av.z : av.y;
    A.y = hi ? av.z        : 1.0f;
    B.x = hi ? 1.0f        : -2.0f * mv.y;
    B.y = hi ? -2.0f * mv.z : mv.y * mv.y;
    acc = __builtin_amdgcn_wmma_f32_16x16x4_f32(false, A, false, B,
                                                (short)0, acc, false, false);
    // ---- step 2: virtual j = 12t+{8..11} ----
    A.x = hi ? av.w        : 1.0f;
    A.y = hi ? 1.0f        : av.w * av.w;
    B.x = hi ? -2.0f * mv.w : mv.z * mv.z;
    B.y = hi ? mv.w * mv.w : 1.0f;
    acc = __builtin_amdgcn_wmma_f32_16x16x4_f32(false, A, false, B,
                                                (short)0, acc, false, false);
  }

  // D layout: VGPR v holds row (c) = v + 8*hi, col (d) = lane&15
  const int col = m;
#pragma unroll
  for (int v = 0; v < 8; ++v) {
    int row = v + (hi ? 8 : 0);
    if (row < NC && col < ND) atomicAdd(&normbuf[col * NC + row], acc[v]);
  }
}

// ---------------------------------------------------------------------------
// Kernel 3: loss[d] = mean_c term1*(term2+term3)  (4 outputs)
// ---------------------------------------------------------------------------
__global__ void loss_kernel(const int* __restrict__ labels,
                            const float* __restrict__ X_weights,
                            const float* __restrict__ sigmas,
                            const float* __restrict__ tao,
                            const float* __restrict__ normbuf,
                            float* __restrict__ out_loss) {
  int d = threadIdx.x;
  if (d >= ND) return;
  float cnt[NC];
#pragma unroll
  for (int c = 0; c < NC; ++c) cnt[c] = X_weights[c];
  for (int b = 0; b < NB; ++b) {
    int l = labels[b];
#pragma unroll
    for (int c = 0; c < NC; ++c) cnt[c] += (l == c) ? 1.0f : 0.0f;
  }
  float t2 = tao[d] * tao[d];
  float accv = 0.0f;
#pragma unroll
  for (int c = 0; c < NC; ++c) {
    float s2 = sigmas[c] * sigmas[c];
    float term1 = s2 / ((t2 + s2) * (t2 + s2));
    float term2 = s2 * normbuf[d * NC + c];
    float term3 = 2.0f * (float)SPA * (t2 * t2 - s2 * s2) / cnt[c];
    accv += term1 * (term2 + term3);
  }
  out_loss[d] = accv * (1.0f / (float)NC);
}

// ---------------------------------------------------------------------------
// Kernel 4: means_theta + log(means_mag + EPS) per (c, s).  w2n == w1n
// (reference computes both from w1).
// ---------------------------------------------------------------------------
__global__ __launch_bounds__(256) void means_kernel(
    const float* __restrict__ x_bins, const float* __restrict__ miu,
    const float* __restrict__ tao, const float* __restrict__ sigmas,
    const float* __restrict__ w1,
    float* __restrict__ meansT, float* __restrict__ logM) {
  int id = blockIdx.x * blockDim.x + threadIdx.x;
  if (id >= NC * SPA) return;
  int c = id >> 14;          // /16384
  int s = id & (SPA - 1);

  float wsum = 0.0f;
  float w1n[ND];
#pragma unroll
  for (int d = 0; d < ND; ++d) { w1n[d] = w1[d] * w1[d]; wsum += w1n[d]; }
#pragma unroll
  for (int d = 0; d < ND; ++d) w1n[d] /= wsum;

  float sig2 = sigmas[c] * sigmas[c];
  float mag = x_bins[(size_t)c * KTOT + SPA + s];   // channel f=1
  float ls_mag = lsig(mag + EPS);

  float mmag = 0.0f, mth = 0.0f;
#pragma unroll
  for (int d = 0; d < ND; ++d) {
    float t2 = tao[d] * tao[d];
    float inv = 1.0f / (sig2 + t2);
    float rt = t2 * inv, rs = sig2 * inv;
    float mu0 = miu[(size_t)d * KTOT + s];          // channel f=0
    float mu1 = miu[(size_t)d * KTOT + SPA + s];    // channel f=1
    mmag += expf((rt * ls_mag + rs * lsig(mu1 + EPS)) * w1n[d]);
    mth += (mag * rt + mu0 * rs) * w1n[d];          // w2n == w1n
  }
  meansT[id] = mth;
  logM[id] = logf(mmag + EPS);
}

// ---------------------------------------------------------------------------
// Kernel 5: dist + min over classes. Block owns a 256-wide s-tile; per-class
// means staged in LDS (20 KB) and reused across all 64 batches.
// ---------------------------------------------------------------------------
__global__ __launch_bounds__(256) void dist_kernel(
    const float* __restrict__ x_LE, const float* __restrict__ meansT,
    const float* __restrict__ logM, const float* __restrict__ weight,
    float* __restrict__ x_out) {
  __shared__ float sT[NC][256];
  __shared__ float sL[NC][256];
  int t = threadIdx.x;
  int s = blockIdx.x * 256 + t;
#pragma unroll
  for (int c = 0; c < NC; ++c) {
    sT[c][t] = meansT[(size_t)c * SPA + s];
    sL[c][t] = logM[(size_t)c * SPA + s];
  }
  __syncthreads();

  float w0 = weight[0] * weight[0];
  float w1 = weight[1] * weight[1];

  for (int b = 0; b < NB; ++b) {
    if (b + 1 < NB) {
      __builtin_prefetch(x_LE + (size_t)(b + 1) * KTOT + s, 0, 1);
      __builtin_prefetch(x_LE + (size_t)(b + 1) * KTOT + SPA + s, 0, 1);
    }
    float x0 = x_LE[(size_t)b * KTOT + s];           // f=0
    float x1 = x_LE[(size_t)b * KTOT + SPA + s];     // f=1
    float lx1 = logf(x1);
    float best = 3.402823466e+38f;
#pragma unroll
    for (int c = 0; c < NC; ++c) {
      float dcur = w0 * fabsf(x0 - sT[c][t]) + w1 * fabsf(lx1 - sL[c][t]);
      best = fminf(best, dcur);
    }
    x_out[(size_t)b * SPA + s] = best;
  }
}

// ---------------------------------------------------------------------------
extern "C" void kernel_launch(void* const* d_in, const int* in_sizes, int n_in,
                              void* d_out, int out_size, void* d_ws, size_t ws_size,
                              hipStream_t stream) {
  const float* x_LE      = (const float*)d_in[0];
  const int*   labels    = (const int*)d_in[1];
  const float* w1        = (const float*)d_in[2];
  // d_in[3] = w2 (unused: reference derives w2n from w1)
  const float* miu       = (const float*)d_in[4];
  const float* tao       = (const float*)d_in[5];
  const float* weight    = (const float*)d_in[6];
  const float* sigmas    = (const float*)d_in[7];
  const float* X_LEs     = (const float*)d_in[8];
  const float* X_weights = (const float*)d_in[9];

  // workspace layout (floats): total 1,114,176 floats ~= 4.5 MB
  float* ws      = (float*)d_ws;
  float* x_bins  = ws;                 // NC*KTOT = 327680
  float* lsA     = ws + 327680;        // NC*KTOT = 327680
  float* lsM     = ws + 655360;        // ND*KTOT = 131072
  float* normbuf = ws + 786432;        // 64 (ND*NC used)
  float* meansT  = ws + 786496;        // NC*SPA = 163840
  float* logM    = ws + 950336;        // NC*SPA = 163840

  float* x_out = (float*)d_out;                    // 64*16384
  float* loss  = (float*)d_out + (size_t)NB * SPA; // 4

  bins_kernel<<<KTOT / 256, 256, 0, stream>>>(x_LE, labels, X_LEs, X_weights,
                                              miu, x_bins, lsA, lsM, normbuf);
  norm_wmma_kernel<<<256, 32, 0, stream>>>(lsA, lsM, normbuf);
  loss_kernel<<<1, 32, 0, stream>>>(labels, X_weights, sigmas, tao, normbuf, loss);
  means_kernel<<<(NC * SPA) / 256, 256, 0, stream>>>(x_bins, miu, tao, sigmas,
                                                     w1, meansT, logM);
  dist_kernel<<<SPA / 256, 256, 0, stream>>>(x_LE, meansT, logM, weight, x_out);
}